// PatientDistillation_77996606095624
// MI455X (gfx1250) — compile-verified
//
#include <hip/hip_runtime.h>
#include <hip/hip_bf16.h>
#include <math.h>
#include <stdint.h>

typedef float v2f __attribute__((ext_vector_type(2)));
typedef float v8f __attribute__((ext_vector_type(8)));
typedef unsigned int v4u __attribute__((ext_vector_type(4)));
typedef unsigned int v8u __attribute__((ext_vector_type(8)));

#define B_N     256
#define D_FEAT  98304          /* 128 * 768 */
#define NKCHUNK 96
#define KCHUNK  (D_FEAT / NKCHUNK)   /* 1024 */

/* TDM-staged inner tiling */
#define KT          64                 /* floats of K staged per TDM tile  */
#define NSTEP       (KCHUNK / KT)      /* 16 double-buffer steps per chunk */
#define LDSROW      68                 /* 64 + 4 pad DWORDs (bank spread)  */
#define TILE_FLOATS (16 * LDSROW)      /* 1088 floats per staged tile      */

/* ------------------------------------------------------------------ */
/* Loss 1 + 2: cross-entropy and temperature-softened KL (tiny).      */
/* ------------------------------------------------------------------ */
__global__ __launch_bounds__(256)
void losses_kernel(const float* __restrict__ t_logits,
                   const float* __restrict__ s_logits,
                   const int*   __restrict__ labels,
                   const int*   __restrict__ temp_p,
                   float*       __restrict__ out) {
    __shared__ float red0[256];
    __shared__ float red1[256];
    const int i = threadIdx.x;
    const float T = (float)(*temp_p);

    float s0 = s_logits[2*i], s1 = s_logits[2*i+1];
    float t0 = t_logits[2*i], t1 = t_logits[2*i+1];

    /* train loss: -log_softmax(s)[label] */
    float m   = fmaxf(s0, s1);
    float lse = m + __logf(__expf(s0 - m) + __expf(s1 - m));
    float logp = ((labels[i] == 0) ? s0 : s1) - lse;
    red0[i] = -logp;

    /* soft loss: KL(softmax(t/T) || softmax(s/T)) */
    float a0 = t0 / T, a1 = t1 / T, b0 = s0 / T, b1 = s1 / T;
    float mt   = fmaxf(a0, a1);
    float lset = mt + __logf(__expf(a0 - mt) + __expf(a1 - mt));
    float lt0 = a0 - lset, lt1 = a1 - lset;
    float p0 = __expf(lt0), p1 = __expf(lt1);
    float ms   = fmaxf(b0, b1);
    float lses = ms + __logf(__expf(b0 - ms) + __expf(b1 - ms));
    float lq0 = b0 - lses, lq1 = b1 - lses;
    red1[i] = p0 * (lt0 - lq0) + p1 * (lt1 - lq1);

    __syncthreads();
    for (int s = 128; s > 0; s >>= 1) {
        if (i < s) { red0[i] += red0[i + s]; red1[i] += red1[i + s]; }
        __syncthreads();
    }
    if (i == 0) {
        out[0] = red0[0] / (float)B_N;
        out[1] = red1[0] / (float)B_N * T * T;
    }
}

/* ------------------------------------------------------------------ */
/* TDM: DMA a 16-row x 64-float fp32 tile (row stride D_FEAT) into    */
/* LDS at byte offset ldsOff, padding each 64-DWORD row with 4 DWORDs */
/* (LDS row stride becomes 68 floats -> conflict-free ds_load_b64).   */
/* D# per CDNA5 ISA; groups 2/3 omitted (2-D).  */
/* ------------------------------------------------------------------ */
__device__ __forceinline__ void tdm_tile_load(const float* gptr, unsigned ldsOff) {
    unsigned long long ga = (unsigned long long)(uintptr_t)gptr;   /* byte addr */

    v4u g0;
    g0[0] = 1u;                                   /* count=1, user, no gather  */
    g0[1] = ldsOff;                               /* lds_addr (bytes)          */
    g0[2] = (unsigned)(ga & 0xFFFFFFFFu);         /* global_addr[31:0]         */
    g0[3] = (unsigned)((ga >> 32) & 0x01FFFFFFu)  /* global_addr[56:32]        */
          | (2u << 30);                           /* type = 2 ("image")        */

    v8u g1;
    g1[0] = (2u << 16)        /* data_size = 4 bytes                  */
          | (1u << 20)        /* pad_enable                           */
          | (5u << 22)        /* pad_interval: 64 DWORDs              */
          | (3u << 25);       /* pad_amount:   4 DWORDs               */
    g1[1] = (unsigned)((D_FEAT & 0xFFFFu) << 16);               /* tensor_dim0[15:0]     */
    g1[2] = (unsigned)(D_FEAT >> 16) | ((B_N & 0xFFFFu) << 16); /* dim0 hi | dim1 lo     */
    g1[3] = ((unsigned)KT << 16);                               /* dim1 hi=0 | tile_dim0 */
    g1[4] = 16u;                                                /* tile_dim1=16, dim2=0  */
    g1[5] = (unsigned)D_FEAT;                                   /* dim0_stride lo32      */
    g1[6] = 0u;                                                 /* stride hi | d1s lo    */
    g1[7] = 0u;

    asm volatile("tensor_load_to_lds %0, %1" :: "s"(g0), "s"(g1) : "memory");
}

/* ------------------------------------------------------------------ */
/* Gram matrix sf @ tf^T via V_WMMA_F32_16X16X4_F32, split-K.         */
/* One wave per 16x16 output tile per K-chunk; grid = (256, 96).      */
/* A/B panels staged into LDS by the Tensor Data Mover, double-       */
/* buffered on TENSORcnt; fragments read with ds_load_b64.            */
/* Diagonal waves (rowTile==colTile) additionally accumulate the row  */
/* squared-norms from the same fragments -> the standalone norms pass */
/* (an extra 200 MB of HBM traffic) is eliminated.                    */
/* ------------------------------------------------------------------ */
__global__ __launch_bounds__(32)
void gram_wmma_kernel(const float* __restrict__ sf,
                      const float* __restrict__ tf,
                      float* __restrict__ gram,
                      float* __restrict__ ns,
                      float* __restrict__ nt) {
    __shared__ float lds[4 * TILE_FLOATS];     /* A0 B0 A1 B1 = 17408 B */

    const int tile    = blockIdx.x;            /* 0..255 */
    const int kc      = blockIdx.y;            /* 0..NKCHUNK-1 */
    const int rowTile = tile >> 4;
    const int colTile = tile & 15;
    const int lane    = threadIdx.x & 31;
    const int r       = lane & 15;
    const int h       = lane >> 4;             /* 0 or 1 */
    const bool diag   = (rowTile == colTile);  /* wave-uniform */

    const float* aG = sf + (size_t)(rowTile * 16) * D_FEAT + (size_t)kc * KCHUNK;
    const float* bG = tf + (size_t)(colTile * 16) * D_FEAT + (size_t)kc * KCHUNK;
    const unsigned ldsBase = (unsigned)(uintptr_t)(&lds[0]);

    /* prime buffer 0 */
    tdm_tile_load(aG, ldsBase);
    tdm_tile_load(bG, ldsBase + TILE_FLOATS * 4u);

    v8f acc = {};
    float na = 0.f, nb = 0.f;
    for (int c = 0; c < NSTEP; ++c) {
        const int cur = c & 1;
        if (c + 1 < NSTEP) {
            const int nxt = cur ^ 1;
            tdm_tile_load(aG + (size_t)(c + 1) * KT,
                          ldsBase + (unsigned)(nxt * 2) * TILE_FLOATS * 4u);
            tdm_tile_load(bG + (size_t)(c + 1) * KT,
                          ldsBase + (unsigned)(nxt * 2 + 1) * TILE_FLOATS * 4u);
            /* in-order TDM: <=2 outstanding means current buffer resident */
            asm volatile("s_wait_tensorcnt 0x2" ::: "memory");
        } else {
            asm volatile("s_wait_tensorcnt 0x0" ::: "memory");
        }

        const int ab = (cur * 2) * TILE_FLOATS;
        const int bb = ab + TILE_FLOATS;
        if (diag) {
#pragma unroll
            for (int kk = 0; kk < KT; kk += 4) {
                v2f a = *(const v2f*)&lds[ab + r * LDSROW + kk + 2 * h];
                v2f b = *(const v2f*)&lds[bb + r * LDSROW + kk + 2 * h];
                na += a.x * a.x + a.y * a.y;       /* co-executes with XDL */
                nb += b.x * b.x + b.y * b.y;
                acc = __builtin_amdgcn_wmma_f32_16x16x4_f32(
                          false, a, false, b, (short)0, acc, false, false);
            }
        } else {
#pragma unroll
            for (int kk = 0; kk < KT; kk += 4) {
                v2f a = *(const v2f*)&lds[ab + r * LDSROW + kk + 2 * h];
                v2f b = *(const v2f*)&lds[bb + r * LDSROW + kk + 2 * h];
                acc = __builtin_amdgcn_wmma_f32_16x16x4_f32(
                          false, a, false, b, (short)0, acc, false, false);
            }
        }
    }

    /* C/D layout: VGPR v -> M = v + 8*(lane>=16), N = lane & 15 */
    const int n     = lane & 15;
    const int mbase = h * 8;
    float* gbase = gram + (size_t)(rowTile * 16) * B_N + colTile * 16 + n;
#pragma unroll
    for (int v = 0; v < 8; ++v) {
        atomicAdd(gbase + (size_t)(mbase + v) * B_N, acc[v]);
    }

    if (diag) {   /* both half-row lanes (h=0,1) contribute to the same row */
        atomicAdd(ns + rowTile * 16 + r, na);
        atomicAdd(nt + colTile * 16 + r, nb);
    }
}

/* ------------------------------------------------------------------ */
/* Per-row: clamped sq-distances, label mask, sum of k smallest / D.  */
/* ------------------------------------------------------------------ */
__global__ __launch_bounds__(256)
void topk_kernel(const float* __restrict__ gram,
                 const float* __restrict__ ns,
                 const float* __restrict__ nt,
                 const int*   __restrict__ labels,
                 const int*   __restrict__ kp,
                 float*       __restrict__ out) {
    __shared__ float d[256];
    const int i = blockIdx.x, j = threadIdx.x;
    const float g  = gram[(size_t)i * B_N + j];
    const float sq = fmaxf(ns[i] + nt[j] - 2.0f * g, 0.0f);
    d[j] = (labels[i] == labels[j]) ? sq : INFINITY;
    __syncthreads();

    if (j == 0) {
        const int k = *kp;
        float s = 0.f;
        for (int it = 0; it < k; ++it) {
            float mn = INFINITY; int mi = -1;
            for (int q = 0; q < B_N; ++q)
                if (d[q] < mn) { mn = d[q]; mi = q; }
            if (mi >= 0 && isfinite(mn)) { s += mn; d[mi] = INFINITY; }
        }
        atomicAdd(out + 2, s / (float)D_FEAT);
    }
}

/* ------------------------------------------------------------------ */
extern "C" void kernel_launch(void* const* d_in, const int* in_sizes, int n_in,
                              void* d_out, int out_size, void* d_ws, size_t ws_size,
                              hipStream_t stream) {
    const float* t_logits   = (const float*)d_in[0];
    const float* s_logits   = (const float*)d_in[1];
    const float* t_features = (const float*)d_in[2];
    const float* s_features = (const float*)d_in[3];
    const int*   labels     = (const int*)d_in[4];
    const int*   kp         = (const int*)d_in[5];
    const int*   temp       = (const int*)d_in[6];

    float* out  = (float*)d_out;
    float* gram = (float*)d_ws;                 /* 256*256 floats  */
    float* ns   = gram + B_N * B_N;             /* 256 floats      */
    float* nt   = ns + B_N;                     /* 256 floats      */

    hipMemsetAsync(d_out, 0, 3 * sizeof(float), stream);
    hipMemsetAsync(d_ws,  0, (B_N * B_N + 2 * B_N) * sizeof(float), stream);

    losses_kernel<<<1, 256, 0, stream>>>(t_logits, s_logits, labels, temp, out);
    gram_wmma_kernel<<<dim3(256, NKCHUNK), 32, 0, stream>>>(s_features, t_features,
                                                            gram, ns, nt);
    topk_kernel<<<256, 256, 0, stream>>>(gram, ns, nt, labels, kp, out);
}